// ContraLoss_62732292325787
// MI455X (gfx1250) — compile-verified
//
#include <hip/hip_runtime.h>

#define NROWS 16384
#define DDIM  128
#define NT    (NROWS / 16)

// 1/(0.2*ln2) = 5*log2(e)
#define SCALE_EXP2 7.213475204444817f
#define LN2F       0.6931471805599453f
#define INV_TEMP   5.0f

typedef __attribute__((ext_vector_type(2))) float v2f;
typedef __attribute__((ext_vector_type(8))) float v8f;
typedef __attribute__((ext_vector_type(4))) int   v4i;

#define AS1 __attribute__((address_space(1)))
#define AS3 __attribute__((address_space(3)))

#if defined(__has_builtin)
#if __has_builtin(__builtin_amdgcn_global_load_async_to_lds_b128) && \
    __has_builtin(__builtin_amdgcn_s_wait_asynccnt)
#define USE_ASYNC 1
#endif
#endif

// ---------------------------------------------------------------------------
// Kernel 1: L2-normalize both views (one wave32 per row) and accumulate
//           sum_n dot(v1n, v2n) into dotAcc via global_atomic_add_f32.
// ---------------------------------------------------------------------------
__global__ __launch_bounds__(256) void nrm_kernel(
    const float* __restrict__ emb, float* __restrict__ v1n,
    float* __restrict__ v2n, float* __restrict__ dotAcc) {
  const int row  = (blockIdx.x * blockDim.x + threadIdx.x) >> 5;
  const int lane = threadIdx.x & 31;
  if (row >= NROWS) return;

  const float4* e1 = (const float4*)(emb + (size_t)row * DDIM);
  const float4* e2 = (const float4*)(emb + (size_t)NROWS * DDIM + (size_t)row * DDIM);
  float4 a = e1[lane];
  float4 b = e2[lane];

  float ss1 = a.x * a.x + a.y * a.y + a.z * a.z + a.w * a.w;
  float ss2 = b.x * b.x + b.y * b.y + b.z * b.z + b.w * b.w;
#pragma unroll
  for (int m = 1; m < 32; m <<= 1) {
    ss1 += __shfl_xor(ss1, m, 32);
    ss2 += __shfl_xor(ss2, m, 32);
  }
  const float i1 = 1.0f / fmaxf(__builtin_sqrtf(ss1), 1e-12f);
  const float i2 = 1.0f / fmaxf(__builtin_sqrtf(ss2), 1e-12f);

  float4 na = make_float4(a.x * i1, a.y * i1, a.z * i1, a.w * i1);
  float4 nb = make_float4(b.x * i2, b.y * i2, b.z * i2, b.w * i2);
  ((float4*)(v1n + (size_t)row * DDIM))[lane] = na;
  ((float4*)(v2n + (size_t)row * DDIM))[lane] = nb;

  float dt = na.x * nb.x + na.y * nb.y + na.z * nb.z + na.w * nb.w;
#pragma unroll
  for (int m = 1; m < 32; m <<= 1) dt += __shfl_xor(dt, m, 32);
  if (lane == 0) unsafeAtomicAdd(dotAcc, dt);
}

// ---------------------------------------------------------------------------
// Kernel 2: fused sim-GEMM + exp-sum. 4 waves/block, each wave owns a 16-row
// strip (A tile resident in 64 VGPRs, ISA 16x4 f32 A layout), sweeps all 1024
// column tiles. B tile double-buffered in LDS via GLOBAL_LOAD_ASYNC_TO_LDS
// (ASYNCcnt), stride 132 -> conflict-free ds_load_b64.
// ---------------------------------------------------------------------------
#define BSTRIDE 132  // 128 + 4 pad: bank = (4*col + ...) % 64, distinct per col

__global__ __launch_bounds__(128) void contra_kernel(
    const float* __restrict__ v1n, const float* __restrict__ v2n,
    const float* __restrict__ dotAcc, float* __restrict__ out) {
  __shared__ float Bt[2][16 * BSTRIDE];

  const int tid  = threadIdx.x;
  const int lane = tid & 31;
  const int wave = tid >> 5;        // 0..3
  const int col  = lane & 15;       // N (or M) index within 16
  const int hi   = lane >> 4;       // K sub-offset selector
  const int rowStart = blockIdx.x * 64 + wave * 16;

  // A tile: rows rowStart..rowStart+15, all 128 K, in registers.
  // Lane layout for V_WMMA_F32_16X16X4_F32 A: VGPR pair {K=4k+2*hi, +1}.
  v2f a[32];
  {
    const float* arow = v1n + (size_t)(rowStart + col) * DDIM + 2 * hi;
#pragma unroll
    for (int k = 0; k < 32; ++k) a[k] = *(const v2f*)(arow + 4 * k);
  }

  float acc[8];
#pragma unroll
  for (int i = 0; i < 8; ++i) acc[i] = 0.0f;

  // Per-tile compute: preload 32 B reg-pairs (pipelined ds_load_2addr_b64),
  // then 32 back-to-back chained fp32 WMMAs, then exp-sum epilogue.
  auto compute_tile = [&](int buf) {
    const float* bbase = &Bt[buf][col * BSTRIDE + 2 * hi];
    v2f b[32];
#pragma unroll
    for (int k = 0; k < 32; ++k) b[k] = *(const v2f*)(bbase + 4 * k);
    v8f c8 = {};
#pragma unroll
    for (int k = 0; k < 32; ++k)
      c8 = __builtin_amdgcn_wmma_f32_16x16x4_f32(
          false, a[k], false, b[k], (short)0, c8, false, false);
#pragma unroll
    for (int i = 0; i < 8; ++i)
      acc[i] += __builtin_amdgcn_exp2f(c8[i] * SCALE_EXP2);
  };

#ifdef USE_ASYNC
  // ---- async double-buffered pipeline: tile j+1 streams while j computes ----
  auto issue_tile = [&](int j, int buf) {
    const float* src = v2n + (size_t)j * 16 * DDIM;
#pragma unroll
    for (int i = 0; i < 4; ++i) {
      const int q = tid + i * 128;  // 0..511 b128 chunks (16 cols x 32 chunks)
      const int c = q >> 5;
      const int w = q & 31;
      AS1 v4i* g =
          (AS1 v4i*)(unsigned long long)(src + q * 4);
      AS3 v4i* l =
          (AS3 v4i*)(unsigned int)(unsigned long long)(&Bt[buf][c * BSTRIDE + w * 4]);
      __builtin_amdgcn_global_load_async_to_lds_b128(g, l, 0, 0);
    }
  };

  issue_tile(0, 0);
  for (int j = 0; j < NT; ++j) {
    if (j + 1 < NT) {
      issue_tile(j + 1, (j + 1) & 1);
      __builtin_amdgcn_s_wait_asynccnt(4);  // older 4 (tile j) have landed
    } else {
      __builtin_amdgcn_s_wait_asynccnt(0);
    }
    __syncthreads();          // tile j visible to all waves
    compute_tile(j & 1);
    __syncthreads();          // protect buf (j&1) before re-issue next iter
  }
#else
  // ---- fallback: synchronous staging through VGPRs ----
  for (int j = 0; j < NT; ++j) {
    const float4* src = (const float4*)(v2n + (size_t)j * 16 * DDIM);
#pragma unroll
    for (int i = 0; i < 4; ++i) {
      const int q = tid + i * 128;
      const int c = q >> 5;
      const int w = q & 31;
      *(float4*)(&Bt[0][c * BSTRIDE + w * 4]) = src[q];
    }
    if (j + 1 < NT)
      __builtin_prefetch((const void*)(src + 512 + tid * 4), 0, 0);
    __syncthreads();
    compute_tile(0);
    __syncthreads();
  }
#endif

  // C layout: lane l holds column (l&15); rows i (lanes 0-15) / 8+i (16-31).
  // Reduce across the 16 lanes of each half-wave -> per-row ttl.
#pragma unroll
  for (int m = 1; m < 16; m <<= 1) {
#pragma unroll
    for (int i = 0; i < 8; ++i) acc[i] += __shfl_xor(acc[i], m, 32);
  }

  if (col == 0) {  // lane 0 (rows r..r+7) and lane 16 (rows r+8..r+15)
    float lsum = 0.0f;
#pragma unroll
    for (int i = 0; i < 8; ++i) lsum += __builtin_amdgcn_logf(acc[i]);
    unsafeAtomicAdd(out, lsum * LN2F);
  }

  // Fold in the positive-pair term exactly once.
  if (blockIdx.x == 0 && tid == 0)
    unsafeAtomicAdd(out, -dotAcc[0] * INV_TEMP);
}

// ---------------------------------------------------------------------------
extern "C" void kernel_launch(void* const* d_in, const int* in_sizes, int n_in,
                              void* d_out, int out_size, void* d_ws, size_t ws_size,
                              hipStream_t stream) {
  (void)in_sizes; (void)n_in; (void)out_size; (void)ws_size;
  const float* emb = (const float*)d_in[0];
  float* out = (float*)d_out;

  float* dotAcc = (float*)d_ws;
  float* v1n = dotAcc + 16;                 // 64B-aligned
  float* v2n = v1n + (size_t)NROWS * DDIM;  // 8 MB each, L2-resident

  (void)hipMemsetAsync(out, 0, sizeof(float), stream);
  (void)hipMemsetAsync(dotAcc, 0, sizeof(float), stream);

  nrm_kernel<<<NROWS / 8, 256, 0, stream>>>(emb, v1n, v2n, dotAcc);
  contra_kernel<<<NROWS / 64, 128, 0, stream>>>(v1n, v2n, dotAcc, out);
}